// Duelling_16673063043609
// MI455X (gfx1250) — compile-verified
//
#include <hip/hip_runtime.h>
#include <hip/hip_bf16.h>

// ---------------------------------------------------------------------------
// GraphSAGE dueling network for MI455X (gfx1250, wave32).
//  - edge aggregation: float atomics resolved in the 192MB L2
//  - dense layers:     V_WMMA_F32_16X16X4_F32 (exact fp32 matrix path),
//                      K-pair-interleaved LDS weights => b64 B-fragment loads,
//                      B fragments batched ahead of the WMMA burst
//  - head:             project-then-aggregate (linearity of segment_sum)
// ---------------------------------------------------------------------------

typedef float v2f __attribute__((ext_vector_type(2)));
typedef float v8f __attribute__((ext_vector_type(8)));

#define N_GRAPHS 64
#define HID 128

// ---------------------------- degree kernels -------------------------------

__global__ void deg_kernel(const long long* __restrict__ dst, float* __restrict__ deg, int nEdges) {
    int e = blockIdx.x * blockDim.x + threadIdx.x;
    if (e < nEdges) atomicAdd(deg + dst[e], 1.0f);
}

__global__ void invdeg_kernel(float* __restrict__ deg, int nNodes) {
    int i = blockIdx.x * blockDim.x + threadIdx.x;
    if (i < nNodes) {
        float d = deg[i];
        deg[i] = (d > 0.0f) ? (1.0f / d) : 0.0f;
    }
}

// ---------------------------- edge scatter ---------------------------------
// One wave per edge. D=128: each lane moves a float4 (coalesced 512B row read).
template <int D>
__global__ void scatter_feat(const float* __restrict__ feat,
                             const long long* __restrict__ src,
                             const long long* __restrict__ dst,
                             float* __restrict__ agg, int nEdges) {
    int e = blockIdx.x * 8 + (threadIdx.x >> 5);
    if (e >= nEdges) return;
    int lane = threadIdx.x & 31;
    long long s = src[e], d = dst[e];
    if (D == 128) {
        float4 v = ((const float4*)(feat + (size_t)s * 128))[lane];
        float* a = agg + (size_t)d * 128 + lane * 4;
        atomicAdd(a + 0, v.x);
        atomicAdd(a + 1, v.y);
        atomicAdd(a + 2, v.z);
        atomicAdd(a + 3, v.w);
    } else {  // D == 32
        float v = feat[(size_t)s * 32 + lane];
        atomicAdd(agg + (size_t)d * 32 + lane, v);
    }
}

__global__ void scatter_scalar(const float* __restrict__ p,
                               const long long* __restrict__ src,
                               const long long* __restrict__ dst,
                               float* __restrict__ aggp, int nEdges) {
    int e = blockIdx.x * blockDim.x + threadIdx.x;
    if (e < nEdges) atomicAdd(aggp + dst[e], p[src[e]]);
}

// ---------------------------- SAGE dense layer -----------------------------
// h_out[m][n] = act( (agg[m]*invdeg[m]) @ Wl + b + h_in[m] @ Wr )
// Block: 256 threads = 8 waves; each wave computes a 16x128 output stripe with
// 8 accumulator tiles of V_WMMA_F32_16X16X4_F32.
//
// LDS weight layout: K-pair interleaved, sW[kp][n][2] = {W[2kp][n], W[2kp+1][n]}
// so a B fragment (v2f, K and K+1 at column n) is ONE aligned ds_load_b64 that
// lands directly in an even VGPR pair. Pair-row stride padded to 288 floats
// (bank shift 32) so the two half-waves (reading adjacent kp) cover disjoint
// bank halves -> conflict-free. Per k4-step all 16 B fragments are fetched
// into registers BEFORE the 16-WMMA burst so LDS latency overlaps the matrix
// pipe instead of serializing load->wait->wmma.
template <int D>
__global__ __launch_bounds__(256) void sage_gemm(
    const float* __restrict__ agg, const float* __restrict__ hin,
    const float* __restrict__ invdeg,
    const float* __restrict__ Wl, const float* __restrict__ Wr,
    const float* __restrict__ bias,
    float* __restrict__ hout, int nNodes, int applyRelu) {
    constexpr int KC = 32;                 // K-chunk
    constexpr int PSTR = 2 * HID + 32;     // padded pair-row stride (floats) = 288
    __shared__ __align__(16) float sWl[(KC / 2) * PSTR];
    __shared__ __align__(16) float sWr[(KC / 2) * PSTR];
    __shared__ float sb[HID];

    const int tid  = threadIdx.x;
    const int wave = tid >> 5;
    const int lane = tid & 31;
    const int l16  = lane & 31 & 15;
    const int m0   = blockIdx.x * 128 + wave * 16;

    if (tid < HID) sb[tid] = bias[tid];

    // A-fragment addressing (16x4 f32 layout): lanes 0-15 hold K={0,1},
    // lanes 16-31 hold K={2,3}; row M = lane&15.
    const int mA = m0 + l16;
    const int mc = (mA < nNodes) ? mA : (nNodes - 1);
    const float idg = invdeg[mc];
    const int koff = (lane < 16) ? 0 : 2;
    const float* aggRow = agg + (size_t)mc * D;
    const float* hinRow = hin + (size_t)mc * D;

    __syncthreads();

    v8f acc[8];
#pragma unroll
    for (int t = 0; t < 8; ++t) {
        float bv = sb[t * 16 + l16];
        acc[t] = (v8f){bv, bv, bv, bv, bv, bv, bv, bv};
    }

    for (int kc = 0; kc < D; kc += KC) {
        __syncthreads();  // previous chunk fully consumed
        // stage Wl/Wr chunk, interleaving K-pairs: (KC/2) pair-rows x 32 col4
        for (int i = tid; i < (KC / 2) * 32; i += 256) {
            int kp = i >> 5;   // pair-row 0..KC/2-1
            int c4 = i & 31;   // float4 column group
            float4 l0 = ((const float4*)(Wl + (size_t)(kc + 2 * kp) * HID))[c4];
            float4 l1 = ((const float4*)(Wl + (size_t)(kc + 2 * kp + 1) * HID))[c4];
            float4* dl = (float4*)(sWl + kp * PSTR + c4 * 8);
            dl[0] = (float4){l0.x, l1.x, l0.y, l1.y};
            dl[1] = (float4){l0.z, l1.z, l0.w, l1.w};
            float4 r0 = ((const float4*)(Wr + (size_t)(kc + 2 * kp) * HID))[c4];
            float4 r1 = ((const float4*)(Wr + (size_t)(kc + 2 * kp + 1) * HID))[c4];
            float4* dr = (float4*)(sWr + kp * PSTR + c4 * 8);
            dr[0] = (float4){r0.x, r1.x, r0.y, r1.y};
            dr[1] = (float4){r0.z, r1.z, r0.w, r1.w};
        }
        // prefetch next chunk while this one computes (global_prefetch_b8)
        if (kc + KC < D) {
            __builtin_prefetch(Wl + (size_t)(kc + KC) * HID + tid * 16, 0, 3);
            __builtin_prefetch(Wr + (size_t)(kc + KC) * HID + tid * 16, 0, 3);
        }
        __syncthreads();

#pragma unroll
        for (int k4 = 0; k4 < KC; k4 += 4) {
            const int kg = kc + k4 + koff;        // global K for A loads
            const int kp = (k4 + koff) >> 1;      // LDS pair-row for B loads
            v2f aAgg = *(const v2f*)(aggRow + kg);
            aAgg.x *= idg;
            aAgg.y *= idg;
            v2f aH = *(const v2f*)(hinRow + kg);
            const float* bl = sWl + kp * PSTR;
            const float* br = sWr + kp * PSTR;
            // batch all 16 B fragments first: 8 back-to-back ds_load_2addr_b64
            // whose latency overlaps the subsequent WMMA burst
            v2f bL[8], bR[8];
#pragma unroll
            for (int t = 0; t < 8; ++t) {
                const int n = t * 16 + l16;
                bL[t] = *(const v2f*)(bl + n * 2);   // {W[k][n], W[k+1][n]}
                bR[t] = *(const v2f*)(br + n * 2);
            }
#pragma unroll
            for (int t = 0; t < 8; ++t) {
                acc[t] = __builtin_amdgcn_wmma_f32_16x16x4_f32(
                    false, aAgg, false, bL[t], (short)0, acc[t], false, false);
                acc[t] = __builtin_amdgcn_wmma_f32_16x16x4_f32(
                    false, aH, false, bR[t], (short)0, acc[t], false, false);
            }
        }
    }

    // C/D layout: VGPR r, lanes 0-15 -> row m0+r, lanes 16-31 -> row m0+r+8,
    // column = t*16 + (lane&15)
    const int rowAdd = (lane < 16) ? 0 : 8;
#pragma unroll
    for (int t = 0; t < 8; ++t) {
#pragma unroll
        for (int r = 0; r < 8; ++r) {
            int row = m0 + r + rowAdd;
            if (row < nNodes) {
                float v = acc[t][r];
                if (applyRelu) v = fmaxf(v, 0.0f);
                hout[(size_t)row * HID + t * 16 + l16] = v;
            }
        }
    }
}

// ---------------------------- head kernels ---------------------------------
// p[n] = embeds[n] . Wa_l ; q[n] = embeds[n] . Wa_r   (one wave per node)
__global__ void head_dots(const float* __restrict__ emb,
                          const float* __restrict__ Wal,
                          const float* __restrict__ War,
                          float* __restrict__ p, float* __restrict__ q, int nNodes) {
    int n = blockIdx.x * 8 + (threadIdx.x >> 5);
    if (n >= nNodes) return;
    int lane = threadIdx.x & 31;
    float4 e4 = ((const float4*)(emb + (size_t)n * HID))[lane];
    float4 wl = ((const float4*)Wal)[lane];
    float4 wr = ((const float4*)War)[lane];
    float pd = e4.x * wl.x + e4.y * wl.y + e4.z * wl.z + e4.w * wl.w;
    float qd = e4.x * wr.x + e4.y * wr.y + e4.z * wr.z + e4.w * wr.w;
#pragma unroll
    for (int off = 16; off > 0; off >>= 1) {
        pd += __shfl_xor(pd, off);
        qd += __shfl_xor(qd, off);
    }
    if (lane == 0) { p[n] = pd; q[n] = qd; }
}

// advantages[n] = 2*tanh(aggp[n]*invdeg[n] + ba + q[n]); also per-graph sums
__global__ void adv_kernel(const float* __restrict__ aggp,
                           const float* __restrict__ invdeg,
                           const float* __restrict__ q,
                           const float* __restrict__ ba,
                           const long long* __restrict__ gidx,
                           float* __restrict__ adv,
                           float* __restrict__ gadvsum,
                           float* __restrict__ gcnt, int nNodes) {
    int n = blockIdx.x * blockDim.x + threadIdx.x;
    if (n >= nNodes) return;
    float a = 2.0f * tanhf(aggp[n] * invdeg[n] + ba[0] + q[n]);
    adv[n] = a;
    long long g = gidx[n];
    atomicAdd(gadvsum + g, a);
    atomicAdd(gcnt + g, 1.0f);
}

// sum-pool embeds per graph (one wave per node, float4 per lane)
__global__ void pool_kernel(const float* __restrict__ emb,
                            const long long* __restrict__ gidx,
                            float* __restrict__ gsum, int nNodes) {
    int n = blockIdx.x * 8 + (threadIdx.x >> 5);
    if (n >= nNodes) return;
    int lane = threadIdx.x & 31;
    float4 v = ((const float4*)(emb + (size_t)n * HID))[lane];
    float* gs = gsum + (size_t)gidx[n] * HID + lane * 4;
    atomicAdd(gs + 0, v.x);
    atomicAdd(gs + 1, v.y);
    atomicAdd(gs + 2, v.z);
    atomicAdd(gs + 3, v.w);
}

// per-graph: value = tanh(gsum . Wv + bv); adv_mean = gadvsum / max(cnt,1)
__global__ void graph_kernel(const float* __restrict__ gsum,
                             const float* __restrict__ Wv,
                             const float* __restrict__ bv,
                             const float* __restrict__ gadvsum,
                             const float* __restrict__ gcnt,
                             float* __restrict__ gval,
                             float* __restrict__ gadvmean) {
    int g = blockIdx.x;
    int lane = threadIdx.x;  // block of 32
    float4 s4 = ((const float4*)(gsum + (size_t)g * HID))[lane];
    float4 w4 = ((const float4*)Wv)[lane];
    float d = s4.x * w4.x + s4.y * w4.y + s4.z * w4.z + s4.w * w4.w;
#pragma unroll
    for (int off = 16; off > 0; off >>= 1) d += __shfl_xor(d, off);
    if (lane == 0) {
        gval[g] = tanhf(d + bv[0]);
        gadvmean[g] = gadvsum[g] / fmaxf(gcnt[g], 1.0f);
    }
}

__global__ void out_kernel(const float* __restrict__ gval,
                           const float* __restrict__ gadvmean,
                           const float* __restrict__ adv,
                           const long long* __restrict__ gidx,
                           float* __restrict__ out, int nNodes) {
    int n = blockIdx.x * blockDim.x + threadIdx.x;
    if (n >= nNodes) return;
    long long g = gidx[n];
    out[n] = tanhf(gval[g] + adv[n] - gadvmean[g]);
}

// ---------------------------- launcher -------------------------------------

extern "C" void kernel_launch(void* const* d_in, const int* in_sizes, int n_in,
                              void* d_out, int out_size, void* d_ws, size_t ws_size,
                              hipStream_t stream) {
    const float* x        = (const float*)d_in[0];
    const long long* ei   = (const long long*)d_in[1];   // [2, E] int64
    const long long* gidx = (const long long*)d_in[2];   // [N] int64
    const float* Wl0 = (const float*)d_in[3];
    const float* b0  = (const float*)d_in[4];
    const float* Wr0 = (const float*)d_in[5];
    const float* Wl1 = (const float*)d_in[6];
    const float* b1  = (const float*)d_in[7];
    const float* Wr1 = (const float*)d_in[8];
    const float* Wl2 = (const float*)d_in[9];
    const float* b2  = (const float*)d_in[10];
    const float* Wr2 = (const float*)d_in[11];
    const float* Wal = (const float*)d_in[12];
    const float* ba  = (const float*)d_in[13];
    const float* War = (const float*)d_in[14];
    const float* Wv  = (const float*)d_in[15];
    const float* bv  = (const float*)d_in[16];
    float* out = (float*)d_out;

    const int N = in_sizes[0] / 32;   // 100000
    const int E = in_sizes[1] / 2;    // 1600000
    const long long* src = ei;
    const long long* dst = ei + E;

    // workspace carve-up (floats)
    float* ws      = (float*)d_ws;
    float* invdeg  = ws;                       // N (deg, then inverted in place)
    float* bufA    = invdeg + N;               // N*128
    float* bufB    = bufA + (size_t)N * HID;   // N*128 (agg)
    float* bufC    = bufB + (size_t)N * HID;   // N*128
    float* p       = bufC + (size_t)N * HID;   // N
    float* q       = p + N;                    // N
    float* aggp    = q + N;                    // N
    float* adv     = aggp + N;                 // N
    float* gsum    = adv + N;                  // 64*128
    float* gcnt    = gsum + N_GRAPHS * HID;    // 64
    float* gadvsum = gcnt + N_GRAPHS;          // 64
    float* gval    = gadvsum + N_GRAPHS;       // 64
    float* gadvmean = gval + N_GRAPHS;         // 64

    const int nodeBlocks = (N + 255) / 256;
    const int edgeBlocks = (E + 255) / 256;
    const int waveNodeBlocks = (N + 7) / 8;   // 8 waves/block, 1 node per wave
    const int waveEdgeBlocks = (E + 7) / 8;
    const int gemmBlocks = (N + 127) / 128;

    // degrees
    hipMemsetAsync(invdeg, 0, (size_t)N * 4, stream);
    deg_kernel<<<edgeBlocks, 256, 0, stream>>>(dst, invdeg, E);
    invdeg_kernel<<<nodeBlocks, 256, 0, stream>>>(invdeg, N);

    // layer 0: in 32 -> 128
    hipMemsetAsync(bufB, 0, (size_t)N * 32 * 4, stream);
    scatter_feat<32><<<waveEdgeBlocks, 256, 0, stream>>>(x, src, dst, bufB, E);
    sage_gemm<32><<<gemmBlocks, 256, 0, stream>>>(bufB, x, invdeg, Wl0, Wr0, b0, bufA, N, 1);

    // layer 1: 128 -> 128
    hipMemsetAsync(bufB, 0, (size_t)N * HID * 4, stream);
    scatter_feat<128><<<waveEdgeBlocks, 256, 0, stream>>>(bufA, src, dst, bufB, E);
    sage_gemm<128><<<gemmBlocks, 256, 0, stream>>>(bufB, bufA, invdeg, Wl1, Wr1, b1, bufC, N, 1);

    // layer 2: 128 -> 128 (no relu) -> embeds in bufA
    hipMemsetAsync(bufB, 0, (size_t)N * HID * 4, stream);
    scatter_feat<128><<<waveEdgeBlocks, 256, 0, stream>>>(bufC, src, dst, bufB, E);
    sage_gemm<128><<<gemmBlocks, 256, 0, stream>>>(bufB, bufC, invdeg, Wl2, Wr2, b2, bufA, N, 0);

    // advantage head: project to scalar first, then aggregate scalars
    head_dots<<<waveNodeBlocks, 256, 0, stream>>>(bufA, Wal, War, p, q, N);
    hipMemsetAsync(aggp, 0, (size_t)N * 4, stream);
    hipMemsetAsync(gsum, 0, (size_t)(N_GRAPHS * HID + 2 * N_GRAPHS) * 4, stream);
    scatter_scalar<<<edgeBlocks, 256, 0, stream>>>(p, src, dst, aggp, E);
    adv_kernel<<<nodeBlocks, 256, 0, stream>>>(aggp, invdeg, q, ba, gidx, adv, gadvsum, gcnt, N);

    // value head: sum-pool + linear + tanh
    pool_kernel<<<waveNodeBlocks, 256, 0, stream>>>(bufA, gidx, gsum, N);
    graph_kernel<<<N_GRAPHS, 32, 0, stream>>>(gsum, Wv, bv, gadvsum, gcnt, gval, gadvmean);

    // final combine
    out_kernel<<<nodeBlocks, 256, 0, stream>>>(gval, gadvmean, adv, gidx, out, N);
}